// Conv_14654428414367
// MI455X (gfx1250) — compile-verified
//
#include <hip/hip_runtime.h>
#include <stdint.h>

// Reference: out = weight[idx] * segment_sum(vals[idx][:,None] * x[cols[idx]], rows[idx], N)
//   N=50000, D=64, R=4, E=800000.
//
// MI455X strategy: atomic/L2-bound scatter (~35MB working set, L2-resident).
//   * One block per 512-edge tile; rows/cols/vals staged into LDS via the
//     gfx1250 async copy path (GLOBAL_LOAD_ASYNC_TO_LDS_B64, ASYNCcnt) --
//     bulk transfers that overlap with co-resident blocks' compute; the
//     inner loop reads metadata as 1-cycle LDS broadcasts (ds_load_b32).
//   * wave32-exact feature tiling: lane l owns features [2l,2l+1] (float2):
//     each x[col] gather is one coalesced 256B burst, each scatter is 2
//     non-returning global_atomic_add_f32 per lane (L2-resident atomics).
//   * 2-edge unroll per wave + global_prefetch_b8 at distance 8 for MLP
//     against the random-gather latency.

#define FEAT_DIM      64
#define TILE          512          // edges per block tile
#define BLOCK_THREADS 256          // 8 waves/block -> 64 edges per wave
#define EDGES_PER_WAVE (TILE / (BLOCK_THREADS / 32))

#ifndef HAVE_ASYNC_LDS
#if defined(__has_builtin)
#if __has_builtin(__builtin_amdgcn_global_load_async_to_lds_b64)
#define HAVE_ASYNC_LDS 1
#endif
#endif
#endif
#ifndef HAVE_ASYNC_LDS
#define HAVE_ASYNC_LDS 0
#endif

// Builtin signature (leaked by round-3 diagnostic):
//   void __builtin_amdgcn_global_load_async_to_lds_b64(
//       v2i global* src, v2i lds* dst, imm int offset, imm int cpol)
typedef int v2i __attribute__((vector_size(8)));
typedef v2i __attribute__((address_space(1)))* gp_v2i;   // global pointer
typedef v2i __attribute__((address_space(3)))* lp_v2i;   // LDS pointer

__device__ __forceinline__ void async_wait_zero() {
#if defined(__has_builtin) && __has_builtin(__builtin_amdgcn_s_wait_asynccnt)
    __builtin_amdgcn_s_wait_asynccnt(0);
#else
    asm volatile("s_wait_asynccnt 0" ::: "memory");
#endif
}

__global__ __launch_bounds__(BLOCK_THREADS)
void spmm_zero_kernel(float4* __restrict__ out4, int n4) {
    int i = blockIdx.x * blockDim.x + threadIdx.x;
    if (i < n4) {
        out4[i] = make_float4(0.0f, 0.0f, 0.0f, 0.0f);
    }
}

__global__ __launch_bounds__(BLOCK_THREADS)
void spmm_coo_tile_kernel(const float* __restrict__ x,
                          const float* __restrict__ weight,
                          const float* __restrict__ edge_vals,
                          const int*   __restrict__ edge_rows,
                          const int*   __restrict__ edge_cols,
                          const int*   __restrict__ idx_ptr,
                          float*       __restrict__ out,
                          int E) {
    __shared__ int   s_rows[TILE];
    __shared__ int   s_cols[TILE];
    __shared__ float s_vals[TILE];

    const int   idx = __builtin_amdgcn_readfirstlane(idx_ptr[0]);
    const float w   = weight[idx];
    const size_t rel = (size_t)idx * (size_t)E;
    const int*   __restrict__ rows = edge_rows + rel;
    const int*   __restrict__ cols = edge_cols + rel;
    const float* __restrict__ vals = edge_vals + rel;

    const int base = blockIdx.x * TILE;
    const int cnt  = min(TILE, E - base);   // edges in this tile (>0 by launch)

    // ---- Stage the metadata tile into LDS (async global->LDS, 8B/thread/array).
    // Partial last tile: clamp the GLOBAL source index so the b64 read stays
    // in-bounds; LDS slots >= cnt hold garbage but are never read below.
    {
        const int t2 = threadIdx.x << 1;          // 2 elements per thread
        const int ei = min(base + t2, E - 2);
#if HAVE_ASYNC_LDS
        __builtin_amdgcn_global_load_async_to_lds_b64(
            (gp_v2i)(rows + ei), (lp_v2i)&s_rows[t2], 0, 0);
        __builtin_amdgcn_global_load_async_to_lds_b64(
            (gp_v2i)(cols + ei), (lp_v2i)&s_cols[t2], 0, 0);
        __builtin_amdgcn_global_load_async_to_lds_b64(
            (gp_v2i)(vals + ei), (lp_v2i)&s_vals[t2], 0, 0);
        async_wait_zero();
#else
        *(int2*)&s_rows[t2]   = *(const int2*)(rows + ei);
        *(int2*)&s_cols[t2]   = *(const int2*)(cols + ei);
        *(float2*)&s_vals[t2] = *(const float2*)(vals + ei);
#endif
    }
    __syncthreads();

    // ---- Process: wave w owns tile edges [w*64, w*64+64) ∩ [0, cnt).
    const int lane2 = (threadIdx.x & 31) << 1;    // 2 floats per lane
    const int wv    = threadIdx.x >> 5;
    const int start = wv * EDGES_PER_WAVE;
    const int end   = min(start + EDGES_PER_WAVE, cnt);

    for (int j = start; j < end; j += 2) {
        const bool hasB = (j + 1) < end;          // wave-uniform guard
        const int  jb   = hasB ? (j + 1) : j;

        // LDS broadcast reads (lane-uniform ds_load_b32).
        const int   colA = s_cols[j];
        const int   rowA = s_rows[j];
        const float vA   = w * s_vals[j];
        const int   colB = s_cols[jb];
        const int   rowB = s_rows[jb];
        const float vB   = w * s_vals[jb];

        // Warm the cache for the gather 8 edges ahead (global_prefetch_b8).
        const int jp = j + 8;
        if (jp < end) {
            __builtin_prefetch(x + (size_t)s_cols[jp] * FEAT_DIM + lane2, 0, 0);
        }

        // Two independent coalesced 256B gathers in flight.
        const float2 xa = *reinterpret_cast<const float2*>(
            x + (size_t)colA * FEAT_DIM + lane2);
        const float2 xb = *reinterpret_cast<const float2*>(
            x + (size_t)colB * FEAT_DIM + lane2);

        // Scatter-add: non-returning hardware f32 atomics (resolve in L2).
        float* oa = out + (size_t)rowA * FEAT_DIM + lane2;
        unsafeAtomicAdd(oa + 0, vA * xa.x);
        unsafeAtomicAdd(oa + 1, vA * xa.y);
        if (hasB) {
            float* ob = out + (size_t)rowB * FEAT_DIM + lane2;
            unsafeAtomicAdd(ob + 0, vB * xb.x);
            unsafeAtomicAdd(ob + 1, vB * xb.y);
        }
    }
}

extern "C" void kernel_launch(void* const* d_in, const int* in_sizes, int n_in,
                              void* d_out, int out_size, void* d_ws, size_t ws_size,
                              hipStream_t stream) {
    const float* x         = (const float*)d_in[0];
    const float* weight    = (const float*)d_in[1];
    const float* edge_vals = (const float*)d_in[2];
    const int*   edge_rows = (const int*)d_in[3];
    const int*   edge_cols = (const int*)d_in[4];
    const int*   idx       = (const int*)d_in[5];
    float*       out       = (float*)d_out;

    const int R = in_sizes[1];                 // 4 relations
    const int E = in_sizes[2] / (R > 0 ? R : 1);

    // Zero the output (harness poisons d_out; must be deterministic per call).
    const int n4 = out_size / 4;               // out_size = N*64, divisible by 4
    const int zblocks = (n4 + 255) / 256;
    spmm_zero_kernel<<<zblocks, 256, 0, stream>>>((float4*)out, n4);

    // One block per 512-edge tile: 1563 blocks * 8 waves ~= 12.5K waves;
    // async metadata staging of one block overlaps compute of co-resident blocks.
    const int blocks = (E + TILE - 1) / TILE;
    spmm_coo_tile_kernel<<<blocks, BLOCK_THREADS, 0, stream>>>(
        x, weight, edge_vals, edge_rows, edge_cols, idx, out, E);
}